// NeuralGWApproximator_24876450578634
// MI455X (gfx1250) — compile-verified
//
#include <hip/hip_runtime.h>
#include <hip/hip_bf16.h>

typedef __attribute__((ext_vector_type(2))) float v2f;
typedef __attribute__((ext_vector_type(8))) float v8f;

#define TILE   128
#define KSLAB  32
#define LDST   (KSLAB + 2)     // 34: bank-conflict-free, even (8B-aligned float2)
#define TEMP_INV 10.0f         // 1 / 0.1
#define MCOLS  6144            // fixed problem size (M == N == 6144)
#define VPT    (MCOLS / 256)   // 24 logits per lane in row_stats

#if defined(__AMDGCN__) && __has_builtin(__builtin_amdgcn_global_load_async_to_lds_b128)
#define HAVE_ASYNC_LDS 1
typedef int v4i __attribute__((vector_size(16)));
typedef __attribute__((address_space(1))) v4i* gas_ptr_t;   // global src
typedef __attribute__((address_space(3))) v4i* las_ptr_t;   // LDS dst
#else
#define HAVE_ASYNC_LDS 0
#endif

// ---------------------------------------------------------------------------
// Generic WMMA f32 GEMM: C[M,N] = act(A[M,K] . B[N,K]^T + bias)
// bias_mode: 0 = none, 1 = per-column vector, 2 = scalar at bias[0]
// act_gelu : 1 = exact GELU
// Tile 128x128, 8 waves (2x4), each wave: 4x2 accumulators of 16x16 f32.
// ---------------------------------------------------------------------------
__global__ __launch_bounds__(256)
void gemm_bt_wmma(const float* __restrict__ A, const float* __restrict__ B,
                  float* __restrict__ C, const float* __restrict__ bias,
                  int M, int N, int K, int bias_mode, int act_gelu)
{
    __shared__ float As[TILE * LDST];
    __shared__ float Bs[TILE * LDST];

    const int tid  = threadIdx.x;
    const int lane = tid & 31;
    const int wave = tid >> 5;
    const int wr   = wave >> 2;   // 0..1  -> 64 rows each
    const int wc   = wave & 3;    // 0..3  -> 32 cols each
    const int half = lane >> 4;   // 0/1
    const int lm   = lane & 15;

    const int tm = blockIdx.y * TILE;
    const int tn = blockIdx.x * TILE;

    v8f acc[4][2];
    const v8f vzero = {0.f,0.f,0.f,0.f,0.f,0.f,0.f,0.f};
#pragma unroll
    for (int mi = 0; mi < 4; ++mi)
#pragma unroll
        for (int ni = 0; ni < 2; ++ni) acc[mi][ni] = vzero;

    for (int k0 = 0; k0 < K; k0 += KSLAB) {
        int ks = K - k0; if (ks > KSLAB) ks = KSLAB;   // 16 or 32, multiple of 4
        // ---- stage A and B tiles (rows tm.. / tn.., cols k0..k0+ks) ----
        const int cpr = ks >> 2;             // float4 per row
        const int nv  = TILE * cpr;          // total float4 per tile
#if HAVE_ASYNC_LDS
        // Direct async global->LDS (no VGPR round-trip), one wait per slab.
        for (int e = tid; e < nv; e += 256) {
            int r  = e / cpr;
            int c4 = (e - r * cpr) << 2;
            __builtin_amdgcn_global_load_async_to_lds_b128(
                (gas_ptr_t)(A + (size_t)(tm + r) * K + k0 + c4),
                (las_ptr_t)&As[r * LDST + c4], 0, 0);
            __builtin_amdgcn_global_load_async_to_lds_b128(
                (gas_ptr_t)(B + (size_t)(tn + r) * K + k0 + c4),
                (las_ptr_t)&Bs[r * LDST + c4], 0, 0);
        }
        asm volatile("s_wait_asynccnt 0x0" ::: "memory");
#else
        for (int e = tid; e < nv; e += 256) {
            int r  = e / cpr;
            int c4 = (e - r * cpr) << 2;
            float4 va = *(const float4*)(A + (size_t)(tm + r) * K + k0 + c4);
            float* da = &As[r * LDST + c4];
            da[0]=va.x; da[1]=va.y; da[2]=va.z; da[3]=va.w;
            float4 vb = *(const float4*)(B + (size_t)(tn + r) * K + k0 + c4);
            float* db = &Bs[r * LDST + c4];
            db[0]=vb.x; db[1]=vb.y; db[2]=vb.z; db[3]=vb.w;
        }
#endif
        __syncthreads();
        // ---- WMMA over the slab, 4 K-steps at a time ----
        for (int kk = 0; kk < ks; kk += 4) {
            const int kp = kk + 2 * half;    // lanes 16-31 hold K pair {k+2,k+3}
            v2f af[4], bf[2];
#pragma unroll
            for (int mi = 0; mi < 4; ++mi) {
                int row = wr * 64 + mi * 16 + lm;
                af[mi] = *(const v2f*)&As[row * LDST + kp];
            }
#pragma unroll
            for (int ni = 0; ni < 2; ++ni) {
                int col = wc * 32 + ni * 16 + lm;
                bf[ni] = *(const v2f*)&Bs[col * LDST + kp];
            }
#pragma unroll
            for (int mi = 0; mi < 4; ++mi)
#pragma unroll
                for (int ni = 0; ni < 2; ++ni)
                    acc[mi][ni] = __builtin_amdgcn_wmma_f32_16x16x4_f32(
                        false, af[mi], false, bf[ni],
                        (short)0, acc[mi][ni], false, false);
        }
        __syncthreads();
    }

    // ---- epilogue: bias + optional exact GELU, store ----
    float bscal = (bias_mode == 2) ? bias[0] : 0.f;
#pragma unroll
    for (int ni = 0; ni < 2; ++ni) {
        const int col = tn + wc * 32 + ni * 16 + lm;
        const float bcol = (bias_mode == 1) ? bias[col] : bscal;
#pragma unroll
        for (int mi = 0; mi < 4; ++mi) {
#pragma unroll
            for (int v = 0; v < 8; ++v) {
                const int row = tm + wr * 64 + mi * 16 + v + 8 * half;
                float x = acc[mi][ni][v] + bcol;
                if (act_gelu) x = 0.5f * x * (1.f + erff(x * 0.70710678118654752f));
                C[(size_t)row * N + col] = x;
            }
        }
    }
}

// WbT[n,j] = Wb[j,n]  (256x256)
__global__ void transpose256(const float* __restrict__ Wb, float* __restrict__ WbT)
{
    int idx = blockIdx.x * 256 + threadIdx.x;
    int j = idx >> 8, n = idx & 255;
    WbT[n * 256 + j] = Wb[j * 256 + n];
}

// Row softmax stats: one block per row; 24 values/lane held in registers
// (fully unrolled so the array cannot spill to scratch) -> single HBM read.
__global__ __launch_bounds__(256)
void row_stats(const float* __restrict__ L, const float* __restrict__ slack,
               float* __restrict__ rmax, float* __restrict__ rinv)
{
    __shared__ float red[256];
    const int row = blockIdx.x;
    const float* Lr = L + (size_t)row * MCOLS;
    float v[VPT];
    float m = -3.4e38f;
#pragma unroll
    for (int t = 0; t < VPT; ++t) {
        v[t] = Lr[threadIdx.x + t * 256];
        m = fmaxf(m, v[t]);
    }
    red[threadIdx.x] = m; __syncthreads();
    for (int s = 128; s > 0; s >>= 1) {
        if (threadIdx.x < s) red[threadIdx.x] = fmaxf(red[threadIdx.x], red[threadIdx.x + s]);
        __syncthreads();
    }
    const float rm = fmaxf(red[0], slack[0]);
    __syncthreads();
    float s = 0.f;
#pragma unroll
    for (int t = 0; t < VPT; ++t) s += expf((v[t] - rm) * TEMP_INV);
    red[threadIdx.x] = s; __syncthreads();
    for (int st = 128; st > 0; st >>= 1) {
        if (threadIdx.x < st) red[threadIdx.x] += red[threadIdx.x + st];
        __syncthreads();
    }
    if (threadIdx.x == 0) {
        float tot = red[0] + expf((slack[0] - rm) * TEMP_INV);
        rmax[row] = rm;
        rinv[row] = 1.f / tot;
    }
}

// Column softmax: online (m,s) partials per 128-row slab (single read, coalesced).
__global__ __launch_bounds__(256)
void col_partial(const float* __restrict__ L, float* __restrict__ pm, float* __restrict__ ps,
                 int Nrows, int Mcols, int rps)
{
    const int c  = blockIdx.x * 256 + threadIdx.x;
    const int r0 = blockIdx.y * rps;
    int r1 = r0 + rps; if (r1 > Nrows) r1 = Nrows;
    float m = -3.4e38f, s = 0.f;
    for (int r = r0; r < r1; ++r) {
        float x = L[(size_t)r * Mcols + c];
        float nm = fmaxf(m, x);
        s = s * expf((m - nm) * TEMP_INV) + expf((x - nm) * TEMP_INV);
        m = nm;
    }
    pm[(size_t)blockIdx.y * Mcols + c] = m;
    ps[(size_t)blockIdx.y * Mcols + c] = s;
}

__global__ __launch_bounds__(256)
void col_merge(const float* __restrict__ pm, const float* __restrict__ ps,
               const float* __restrict__ slack,
               float* __restrict__ cmax, float* __restrict__ cinv, int Mcols, int nSlab)
{
    const int c = blockIdx.x * 256 + threadIdx.x;
    float m = -3.4e38f, s = 0.f;
    for (int k = 0; k < nSlab; ++k) {
        float mk = pm[(size_t)k * Mcols + c];
        float sk = ps[(size_t)k * Mcols + c];
        float nm = fmaxf(m, mk);
        s = s * expf((m - nm) * TEMP_INV) + sk * expf((mk - nm) * TEMP_INV);
        m = nm;
    }
    const float sl = slack[0];
    float nm = fmaxf(m, sl);
    s = s * expf((m - nm) * TEMP_INV) + expf((sl - nm) * TEMP_INV);
    cmax[c] = nm;
    cinv[c] = 1.f / s;
}

// pi = 0.5 * (row_softmax + col_softmax).
// Last read of logits (non-temporal / last-use), streaming store of pi (NT):
// keeps the L2 free for data that is actually re-read.
__global__ __launch_bounds__(256)
void final_combine(const float* __restrict__ L,
                   const float* __restrict__ rmax, const float* __restrict__ rinv,
                   const float* __restrict__ cmax, const float* __restrict__ cinv,
                   float* __restrict__ pi, int Mcols)
{
    const int bpr = Mcols >> 8;
    const int row = blockIdx.x / bpr;
    const int j   = (blockIdx.x - row * bpr) * 256 + threadIdx.x;
    const size_t idx = (size_t)row * Mcols + j;
    const float x  = __builtin_nontemporal_load(&L[idx]);
    const float pr = expf((x - rmax[row]) * TEMP_INV) * rinv[row];
    const float pc = expf((x - cmax[j]) * TEMP_INV) * cinv[j];
    __builtin_nontemporal_store(0.5f * (pr + pc), &pi[idx]);
}

extern "C" void kernel_launch(void* const* d_in, const int* in_sizes, int n_in,
                              void* d_out, int out_size, void* d_ws, size_t ws_size,
                              hipStream_t stream)
{
    (void)in_sizes; (void)n_in; (void)out_size; (void)ws_size;
    const float* Xq  = (const float*)d_in[0];
    const float* Xr  = (const float*)d_in[1];
    const float* W1  = (const float*)d_in[2];
    const float* b1  = (const float*)d_in[3];
    const float* W2  = (const float*)d_in[4];
    const float* b2  = (const float*)d_in[5];
    const float* Wb  = (const float*)d_in[6];
    const float* bb  = (const float*)d_in[7];
    const float* slq = (const float*)d_in[8];
    const float* slr = (const float*)d_in[9];

    const int N = 6144, M = 6144, H = 256, DIN = 16;
    const int SLABS = 48, RPS = 128;

    float* pi     = (float*)d_out;
    float* logits = (float*)d_out + (size_t)N * M;

    char* w = (char*)d_ws;
    float* Abuf = (float*)w; w += (size_t)N * H * 4;
    float* Hq   = (float*)w; w += (size_t)N * H * 4;
    float* Hr   = (float*)w; w += (size_t)M * H * 4;
    float* G    = (float*)w; w += (size_t)N * H * 4;
    float* WbT  = (float*)w; w += (size_t)H * H * 4;
    float* rmx  = (float*)w; w += (size_t)N * 4;
    float* rin  = (float*)w; w += (size_t)N * 4;
    float* cmx  = (float*)w; w += (size_t)M * 4;
    float* cin  = (float*)w; w += (size_t)M * 4;
    float* pm   = (float*)w; w += (size_t)SLABS * M * 4;
    float* ps   = (float*)w; w += (size_t)SLABS * M * 4;

    dim3 blk(256);

    // compress q: a = gelu(Xq W1^T + b1); Hq = a W2^T + b2
    gemm_bt_wmma<<<dim3(H/128, N/128), blk, 0, stream>>>(Xq,  W1, Abuf, b1, N, H, DIN, 1, 1);
    gemm_bt_wmma<<<dim3(H/128, N/128), blk, 0, stream>>>(Abuf, W2, Hq,  b2, N, H, H,   1, 0);
    // compress r
    gemm_bt_wmma<<<dim3(H/128, M/128), blk, 0, stream>>>(Xr,  W1, Abuf, b1, M, H, DIN, 1, 1);
    gemm_bt_wmma<<<dim3(H/128, M/128), blk, 0, stream>>>(Abuf, W2, Hr,  b2, M, H, H,   1, 0);
    // G = Hq @ Wb  (via WbT so it is A . B^T)
    transpose256<<<dim3((H*H)/256), blk, 0, stream>>>(Wb, WbT);
    gemm_bt_wmma<<<dim3(H/128, N/128), blk, 0, stream>>>(Hq, WbT, G, nullptr, N, H, H, 0, 0);
    // logits = G @ Hr^T + b_bil
    gemm_bt_wmma<<<dim3(M/128, N/128), blk, 0, stream>>>(G, Hr, logits, bb, N, M, H, 2, 0);

    // softmax stats + combine
    row_stats  <<<dim3(N),            blk, 0, stream>>>(logits, slq, rmx, rin);
    col_partial<<<dim3(M/256, SLABS), blk, 0, stream>>>(logits, pm, ps, N, M, RPS);
    col_merge  <<<dim3(M/256),        blk, 0, stream>>>(pm, ps, slr, cmx, cin, M, SLABS);
    final_combine<<<dim3(N * (M/256)), blk, 0, stream>>>(logits, rmx, rin, cmx, cin, pi, M);
}